// SimpleNetworkLayer_11209864642665
// MI455X (gfx1250) — compile-verified
//
#include <hip/hip_runtime.h>
#include <hip/hip_bf16.h>

#define N_NODES 50000
#define N_EDGES 400000

typedef __attribute__((ext_vector_type(2))) float v2f;
typedef __attribute__((ext_vector_type(8))) float v8f;
typedef __attribute__((ext_vector_type(4))) unsigned v4u;
typedef __attribute__((ext_vector_type(8))) int v8i;
typedef __attribute__((ext_vector_type(4))) int v4i;

__device__ __forceinline__ float fast_sigmoid(float x) {
    // 1/(1+exp(-x)) via v_exp_f32 + v_rcp_f32 (avoids IEEE div fixup chains)
    return __builtin_amdgcn_rcpf(1.0f + __builtin_amdgcn_exp2f(-x * 1.442695040888963f));
}
__device__ __forceinline__ float silu_f(float x) { return x * fast_sigmoid(x); }

static __device__ __forceinline__ v8f vzero8() {
    v8f z = {0.f,0.f,0.f,0.f,0.f,0.f,0.f,0.f};
    return z;
}

#define WMMA_F32(A,B,C) __builtin_amdgcn_wmma_f32_16x16x4_f32(false,(A),false,(B),(short)0,(C),false,false)

// ---------------------------------------------------------------------------
// TDM: DMA a contiguous run of n f32 from global into LDS (1-row 2D tile).
// D# packing per CDNA5 ISA (6-arg clang-23 builtin).
// ---------------------------------------------------------------------------
__device__ __forceinline__ void tdm_load_1d(void* lds_dst, const void* gsrc, unsigned n) {
    const unsigned lds_addr = (unsigned)(size_t)lds_dst;          // flat LDS addr truncates to LDS offset
    const unsigned long long ga = (unsigned long long)(size_t)gsrc;
    v4u g0;
    g0.x = 1u;                                                    // count=1, user descriptor
    g0.y = lds_addr;                                              // bits 63:32 lds_addr
    g0.z = (unsigned)(ga & 0xFFFFFFFFull);                        // global_addr lo
    g0.w = (unsigned)((ga >> 32) & 0x01FFFFFFull) | (2u << 30);   // global_addr hi | type=2
    v8i g1;
    g1[0] = (int)(2u << 16);                                      // wg_mask=0 | data_size=2 (4B)
    g1[1] = (int)((n & 0xFFFFu) << 16);                           // tensor_dim0[15:0] @ bits63:48
    g1[2] = (int)(((n >> 16) & 0xFFFFu) | (1u << 16));            // tensor_dim0[31:16] | tensor_dim1=1
    g1[3] = (int)((n & 0xFFFFu) << 16);                           // tensor_dim1 hi=0 | tile_dim0=n
    g1[4] = 1;                                                    // tile_dim1=1 | tile_dim2=0
    g1[5] = (int)n;                                               // tensor_dim0_stride lo32
    g1[6] = 0;                                                    // stride hi | dim1_stride lo
    g1[7] = 0;
    const v4i z4 = {0,0,0,0};                                     // groups 2/3 unused (<=2D)
    const v8i z8 = {0,0,0,0,0,0,0,0};
    __builtin_amdgcn_tensor_load_to_lds(g0, g1, z4, z4, z8, 0);
}

// ---------------------------------------------------------------------------
// Zero the aggregation workspace (agg_s N*64 | agg_v N*192 | cnt N)
// ---------------------------------------------------------------------------
__global__ void zero_ws_kernel(float* __restrict__ p, int n) {
    int i = blockIdx.x * blockDim.x + threadIdx.x;
    int stride = gridDim.x * blockDim.x;
    for (; i < n; i += stride) p[i] = 0.0f;
}

// ---------------------------------------------------------------------------
// Edge kernel: 4 waves/block, one 16-edge tile per wave (400000 = 6250*64,
// so no tail anywhere). Edge MLP runs on v_wmma_f32_16x16x4_f32:
//   h1 A-frags generated in-lane, h2 = silu(h1@w2+b2) staged via LDS,
//   scal = h2@w3+b3 staged via LDS, then 2-lanes-per-edge atomic scatter.
// w2/w3 are DMA'd into LDS with tensor_load_to_lds.
// ---------------------------------------------------------------------------
__global__ __launch_bounds__(128) void edge_kernel(
    const float* __restrict__ node_scalars, const float* __restrict__ node_vectors,
    const float* __restrict__ sh,           const float* __restrict__ norm,
    const float* __restrict__ w1, const float* __restrict__ b1,
    const float* __restrict__ w2, const float* __restrict__ b2,
    const float* __restrict__ w3, const float* __restrict__ b3,
    const int* __restrict__ senders, const int* __restrict__ receivers,
    float* __restrict__ agg_s, float* __restrict__ agg_v, float* __restrict__ cnt)
{
    __shared__ float sw2[64 * 64];        // 16KB
    __shared__ float sw3[64 * 128];       // 32KB
    __shared__ float h2s[4][16 * 64];     // 16KB  per-wave h2 tiles
    __shared__ float scals[4][16 * 128];  // 32KB  per-wave scal tiles

    // --- async tensor DMA of the shared B-operands (wave 0 only) ---
    if (threadIdx.x == 0) {
        tdm_load_1d(sw2, w2, 64 * 64);
        tdm_load_1d(sw3, w3, 64 * 128);
        __builtin_amdgcn_s_wait_tensorcnt(0);
    }
    __syncthreads();

    const int tid  = threadIdx.x;
    const int wid  = tid >> 5;
    const int lane = tid & 31;
    const int lrow = lane & 15;
    const int hi   = lane >> 4;

    const int ebase = (blockIdx.x * 4 + wid) * 16;
    const int e     = ebase + lrow;            // this lane's edge row

    // --- h1 A-fragments: lane generates exactly its A-layout elements ---
    const float nrm = norm[e];
    v2f aH[16];
    #pragma unroll
    for (int kk = 0; kk < 16; ++kk) {
        const int k0 = 4*kk + 2*hi;
        aH[kk].x = silu_f(fmaf(nrm, w1[k0],   b1[k0]));
        aH[kk].y = silu_f(fmaf(nrm, w1[k0+1], b1[k0+1]));
    }

    // --- h2 = silu(h1 @ w2 + b2), 4 n-tiles of 16 ---
    float* h2w = h2s[wid];
    for (int t2 = 0; t2 < 4; ++t2) {
        v8f acc = vzero8();
        #pragma unroll
        for (int kk = 0; kk < 16; ++kk) {
            const int k0 = 4*kk + 2*hi;
            v2f b; b.x = sw2[k0*64 + 16*t2 + lrow]; b.y = sw2[(k0+1)*64 + 16*t2 + lrow];
            acc = WMMA_F32(aH[kk], b, acc);
        }
        const float bias = b2[16*t2 + lrow];
        #pragma unroll
        for (int j = 0; j < 8; ++j) {
            const int row = j + 8*hi;
            h2w[row*64 + 16*t2 + lrow] = silu_f(acc[j] + bias);
        }
    }
    asm volatile("" ::: "memory");

    // --- scal = h2 @ w3 + b3, 8 n-tiles of 16 ---
    v2f aS[16];
    #pragma unroll
    for (int kk = 0; kk < 16; ++kk) {
        const int k0 = 4*kk + 2*hi;
        aS[kk].x = h2w[lrow*64 + k0];
        aS[kk].y = h2w[lrow*64 + k0 + 1];
    }
    float* scw = scals[wid];
    for (int t3 = 0; t3 < 8; ++t3) {
        v8f acc = vzero8();
        #pragma unroll
        for (int kk = 0; kk < 16; ++kk) {
            const int k0 = 4*kk + 2*hi;
            v2f b; b.x = sw3[k0*128 + 16*t3 + lrow]; b.y = sw3[(k0+1)*128 + 16*t3 + lrow];
            acc = WMMA_F32(aS[kk], b, acc);
        }
        const float bias = b3[16*t3 + lrow];
        #pragma unroll
        for (int j = 0; j < 8; ++j) {
            const int row = j + 8*hi;
            scw[row*128 + 16*t3 + lrow] = acc[j] + bias;
        }
    }
    asm volatile("" ::: "memory");

    // --- scatter: 2 lanes per edge (hi selects m-halves), L2-resident atomics ---
    const int s = senders[e];
    const int r = receivers[e];
    const float y0  = sh[e*4 + 0];
    const float y1x = sh[e*4 + 1];
    const float y1y = sh[e*4 + 2];
    const float y1z = sh[e*4 + 3];
    if (hi == 0) unsafeAtomicAdd(&cnt[r], 1.0f);

    const float* srow = node_scalars + (size_t)s * 32;
    const float* vrow = node_vectors + (size_t)s * 96;
    float* aggsr = agg_s + (size_t)r * 64;
    float* aggvr = agg_v + (size_t)r * 192;
    const float* myscal = scw + lrow * 128;
    const float inv_sqrt3 = 0.5773502691896258f;

    #pragma unroll 1
    for (int mm = 0; mm < 16; ++mm) {
        const int m = mm + 16*hi;
        const float sc0 = myscal[m], sc1 = myscal[32+m], sc2 = myscal[64+m], sc3 = myscal[96+m];
        const float se = srow[m];
        const float v0 = vrow[m*3+0], v1 = vrow[m*3+1], v2 = vrow[m*3+2];
        unsafeAtomicAdd(&aggsr[m],      y0 * se * sc0);
        const float dv = (y1x*v0 + y1y*v1 + y1z*v2) * inv_sqrt3;
        unsafeAtomicAdd(&aggsr[32 + m], dv * sc1);
        unsafeAtomicAdd(&aggvr[m*3+0], y0 * v0 * sc2);
        unsafeAtomicAdd(&aggvr[m*3+1], y0 * v1 * sc2);
        unsafeAtomicAdd(&aggvr[m*3+2], y0 * v2 * sc2);
        const float t3v = se * sc3;
        unsafeAtomicAdd(&aggvr[(32+m)*3+0], y1x * t3v);
        unsafeAtomicAdd(&aggvr[(32+m)*3+1], y1y * t3v);
        unsafeAtomicAdd(&aggvr[(32+m)*3+2], y1z * t3v);
    }
}

// ---------------------------------------------------------------------------
// Node kernel: each wave owns a 16-node tile; all linears via
// v_wmma_f32_16x16x4_f32. 4 waves / block.
// ---------------------------------------------------------------------------
__global__ __launch_bounds__(128) void node_kernel(
    const float* __restrict__ node_scalars, const float* __restrict__ node_vectors,
    const float* __restrict__ l1ws, const float* __restrict__ l1wv,
    const float* __restrict__ l2ws, const float* __restrict__ l2wv,
    const float* __restrict__ agg_s, const float* __restrict__ agg_v,
    const float* __restrict__ cnt,  float* __restrict__ out)
{
    __shared__ float stage[4][16 * 16];   // per-wave C->A transpose tile
    __shared__ float gateb[4][16 * 64];   // per-wave gates [row][ch]

    const int tid  = threadIdx.x;
    const int wid  = tid >> 5;
    const int lane = tid & 31;
    const int lrow = lane & 15;
    const int hi   = lane >> 4;

    const int base = (blockIdx.x * 4 + wid) * 16;
    if (base >= N_NODES) return;   // wave-uniform exit

    const int  nodeA  = base + lrow;
    const bool validA = (nodeA < N_NODES);
    const int  ndA    = validA ? nodeA : (N_NODES - 1);

    const float s160 = 0.07905694150420949f;  // 1/sqrt(160)
    const float s96  = 0.10206207261596575f;  // 1/sqrt(96)

    const float cv    = cnt[ndA];
    const float rdA   = validA ? (0.125f / fmaxf(cv, 1.0f)) : 0.0f;  // (1/denom)*(1/sqrt(64))
    const float sA160 = validA ? s160 : 0.0f;
    const float sA96  = validA ? s96  : 0.0f;

    const float* aggsRow = agg_s        + (size_t)ndA * 64;
    const float* aggvRow = agg_v        + (size_t)ndA * 192;
    const float* nsRow   = node_scalars + (size_t)ndA * 32;
    const float* nvRow   = node_vectors + (size_t)ndA * 96;

    float* st = stage[wid];
    float* gb = gateb[wid];

    // ---------- Phase 1: gates = sigmoid(x_s[:,128:192]) ----------
    for (int t = 0; t < 4; ++t) {
        v8f acc = vzero8();
        const int n0 = 128 + 16 * t;
        #pragma unroll
        for (int kk = 0; kk < 16; ++kk) {
            const int k0 = 4*kk + 2*hi;
            v2f a; a.x = aggsRow[k0] * rdA;            a.y = aggsRow[k0+1] * rdA;
            v2f b; b.x = l1ws[k0*192 + n0 + lrow];     b.y = l1ws[(k0+1)*192 + n0 + lrow];
            acc = WMMA_F32(a, b, acc);
        }
        #pragma unroll
        for (int j = 0; j < 8; ++j) {
            const int row = j + 8*hi;
            gb[row*64 + 16*t + lrow] = fast_sigmoid(acc[j]);
        }
    }
    asm volatile("" ::: "memory");

    // ---------- Phase 2+3: o_s = c_s @ lin2_w_s / sqrt(160) ----------
    v8f o0 = vzero8(), o1 = vzero8();
    for (int t = 0; t < 8; ++t) {
        v8f acc = vzero8();
        const int n0 = 16 * t;
        #pragma unroll
        for (int kk = 0; kk < 16; ++kk) {
            const int k0 = 4*kk + 2*hi;
            v2f a; a.x = aggsRow[k0] * rdA;            a.y = aggsRow[k0+1] * rdA;
            v2f b; b.x = l1ws[k0*192 + n0 + lrow];     b.y = l1ws[(k0+1)*192 + n0 + lrow];
            acc = WMMA_F32(a, b, acc);
        }
        asm volatile("" ::: "memory");
        #pragma unroll
        for (int j = 0; j < 8; ++j) {
            const int row = j + 8*hi;
            st[row*16 + lrow] = silu_f(acc[j]) * s160;   // act_s, pre-scaled
        }
        asm volatile("" ::: "memory");
        #pragma unroll
        for (int kk = 0; kk < 4; ++kk) {
            const int kloc = 4*kk + 2*hi;
            v2f a; a.x = st[lrow*16 + kloc];  a.y = st[lrow*16 + kloc + 1];
            const int kg = 16*t + kloc;
            v2f b0; b0.x = l2ws[kg*32 + lrow];      b0.y = l2ws[(kg+1)*32 + lrow];
            v2f b1; b1.x = l2ws[kg*32 + 16 + lrow]; b1.y = l2ws[(kg+1)*32 + 16 + lrow];
            o0 = WMMA_F32(a, b0, o0);
            o1 = WMMA_F32(a, b1, o1);
        }
    }
    // c_s tail: raw node_scalars rows 128..159
    #pragma unroll
    for (int kk = 0; kk < 8; ++kk) {
        const int kc = 4*kk + 2*hi;                  // node_scalars col 0..31
        v2f a; a.x = nsRow[kc] * sA160;  a.y = nsRow[kc+1] * sA160;
        const int kg = 128 + kc;
        v2f b0; b0.x = l2ws[kg*32 + lrow];      b0.y = l2ws[(kg+1)*32 + lrow];
        v2f b1; b1.x = l2ws[kg*32 + 16 + lrow]; b1.y = l2ws[(kg+1)*32 + 16 + lrow];
        o0 = WMMA_F32(a, b0, o0);
        o1 = WMMA_F32(a, b1, o1);
    }
    #pragma unroll
    for (int j = 0; j < 8; ++j) {
        const int row = j + 8*hi;
        const int node = base + row;
        if (node < N_NODES) {
            out[(size_t)node*128 + lrow]      = o0[j];
            out[(size_t)node*128 + 16 + lrow] = o1[j];
        }
    }

    // ---------- Phase 4+5: o_v per component ----------
    for (int i = 0; i < 3; ++i) {
        v8f ov0 = vzero8(), ov1 = vzero8();
        for (int v = 0; v < 4; ++v) {
            v8f acc = vzero8();
            #pragma unroll
            for (int kk = 0; kk < 16; ++kk) {
                const int k0 = 4*kk + 2*hi;
                v2f a; a.x = aggvRow[k0*3 + i] * rdA;      a.y = aggvRow[(k0+1)*3 + i] * rdA;
                v2f b; b.x = l1wv[k0*64 + 16*v + lrow];    b.y = l1wv[(k0+1)*64 + 16*v + lrow];
                acc = WMMA_F32(a, b, acc);
            }
            asm volatile("" ::: "memory");
            #pragma unroll
            for (int j = 0; j < 8; ++j) {
                const int row = j + 8*hi;
                const float g = gb[row*64 + 16*v + lrow];
                st[row*16 + lrow] = acc[j] * g * s96;      // gated x_v, pre-scaled
            }
            asm volatile("" ::: "memory");
            #pragma unroll
            for (int kk = 0; kk < 4; ++kk) {
                const int kloc = 4*kk + 2*hi;
                v2f a; a.x = st[lrow*16 + kloc];  a.y = st[lrow*16 + kloc + 1];
                const int kg = 16*v + kloc;
                v2f b0; b0.x = l2wv[kg*32 + lrow];      b0.y = l2wv[(kg+1)*32 + lrow];
                v2f b1; b1.x = l2wv[kg*32 + 16 + lrow]; b1.y = l2wv[(kg+1)*32 + 16 + lrow];
                ov0 = WMMA_F32(a, b0, ov0);
                ov1 = WMMA_F32(a, b1, ov1);
            }
        }
        // c_v tail: raw node_vectors rows 64..95
        #pragma unroll
        for (int kk = 0; kk < 8; ++kk) {
            const int kc = 4*kk + 2*hi;                // node_vectors ch 0..31
            v2f a; a.x = nvRow[kc*3 + i] * sA96;  a.y = nvRow[(kc+1)*3 + i] * sA96;
            const int kg = 64 + kc;
            v2f b0; b0.x = l2wv[kg*32 + lrow];      b0.y = l2wv[(kg+1)*32 + lrow];
            v2f b1; b1.x = l2wv[kg*32 + 16 + lrow]; b1.y = l2wv[(kg+1)*32 + 16 + lrow];
            ov0 = WMMA_F32(a, b0, ov0);
            ov1 = WMMA_F32(a, b1, ov1);
        }
        #pragma unroll
        for (int j = 0; j < 8; ++j) {
            const int row = j + 8*hi;
            const int node = base + row;
            if (node < N_NODES) {
                out[(size_t)node*128 + 32 + lrow*3 + i]        = ov0[j];
                out[(size_t)node*128 + 32 + (16 + lrow)*3 + i] = ov1[j];
            }
        }
    }
}

// ---------------------------------------------------------------------------
extern "C" void kernel_launch(void* const* d_in, const int* in_sizes, int n_in,
                              void* d_out, int out_size, void* d_ws, size_t ws_size,
                              hipStream_t stream) {
    const float* node_scalars = (const float*)d_in[0];
    const float* node_vectors = (const float*)d_in[1];
    const float* sh           = (const float*)d_in[2];
    const float* norm         = (const float*)d_in[3];
    const float* mlp_w1       = (const float*)d_in[4];
    const float* mlp_b1       = (const float*)d_in[5];
    const float* mlp_w2       = (const float*)d_in[6];
    const float* mlp_b2       = (const float*)d_in[7];
    const float* mlp_w3       = (const float*)d_in[8];
    const float* mlp_b3       = (const float*)d_in[9];
    const float* lin1_w_s     = (const float*)d_in[10];
    const float* lin1_w_v     = (const float*)d_in[11];
    const float* lin2_w_s     = (const float*)d_in[12];
    const float* lin2_w_v     = (const float*)d_in[13];
    const int*   senders      = (const int*)d_in[14];
    const int*   receivers    = (const int*)d_in[15];

    float* ws    = (float*)d_ws;
    float* agg_s = ws;                                  // N*64
    float* agg_v = agg_s + (size_t)N_NODES * 64;        // N*192
    float* cnt   = agg_v + (size_t)N_NODES * 192;       // N

    const int ztotal = N_NODES * 257;
    zero_ws_kernel<<<dim3(2048), dim3(256), 0, stream>>>(ws, ztotal);

    edge_kernel<<<dim3(N_EDGES / 64), dim3(128), 0, stream>>>(
        node_scalars, node_vectors, sh, norm,
        mlp_w1, mlp_b1, mlp_w2, mlp_b2, mlp_w3, mlp_b3,
        senders, receivers, agg_s, agg_v, cnt);

    node_kernel<<<dim3((N_NODES + 63) / 64), dim3(128), 0, stream>>>(
        node_scalars, node_vectors,
        lin1_w_s, lin1_w_v, lin2_w_s, lin2_w_v,
        agg_s, agg_v, cnt, (float*)d_out);
}